// PL_40132174414419
// MI455X (gfx1250) — compile-verified
//
#include <hip/hip_runtime.h>
#include <hip/hip_bf16.h>

// Persistence landscape: per (b,d,c) slice, top-2 over N=1024 triangle values
// at each of T=100 sample times. B=64, D=2, C=8, N=1024, T=100, K=2.
// Input : (64,2,8,1024,2) f32  = 2,097,152 floats (8 MB)
// Output: (64,2,2,800)    f32  =   204,800 floats

#define PL_T     100
#define PL_N     1024
#define PL_C     8
#define PL_PTS_PER_LANE 32   // N / 32 lanes

#define AS1 __attribute__((address_space(1)))
#define AS3 __attribute__((address_space(3)))

typedef int v4i __attribute__((vector_size(16)));

#if __has_builtin(__builtin_amdgcn_global_load_async_to_lds_b128) && \
    __has_builtin(__builtin_amdgcn_s_wait_asynccnt)
#define PL_USE_ASYNC 1
#else
#define PL_USE_ASYNC 0
#endif

__launch_bounds__(256, 4)
__global__ void pl_topk_kernel(const float* __restrict__ diag,
                               float* __restrict__ out) {
  __shared__ __align__(16) float2 pts_lds[PL_N];

  const int s   = blockIdx.x;        // slice = (b*D + d)*C + c
  const int tid = threadIdx.x;
  const float* gsrc = diag + (size_t)s * (PL_N * 2);

  // ---- Stage the slice (8 KB) into LDS ---------------------------------
#if PL_USE_ASYNC
  {
    AS1 v4i* g = (AS1 v4i*)(uintptr_t)(const void*)gsrc;
    AS3 v4i* l = (AS3 v4i*)(uint32_t)(uintptr_t)(void*)pts_lds;
    // 256 lanes x 16 B x 2 = 8192 B, tracked on ASYNCcnt
    __builtin_amdgcn_global_load_async_to_lds_b128(g + tid,       l + tid,       0, 0);
    __builtin_amdgcn_global_load_async_to_lds_b128(g + 256 + tid, l + 256 + tid, 0, 0);
    __builtin_amdgcn_s_wait_asynccnt(0);   // my wave's async writes landed
  }
  __syncthreads();                          // everyone else's too
#else
  {
    const float4* g4 = (const float4*)gsrc;
    float4*       l4 = (float4*)pts_lds;
    l4[tid]       = g4[tid];
    l4[tid + 256] = g4[tid + 256];
  }
  __syncthreads();
#endif

  const int lane = tid & 31;
  const int wave = tid >> 5;

  // ---- Register-stage this lane's 32 strided points (ds_load_b64, no
  //      bank conflicts: lane l hits banks 2l, 2l+1) ---------------------
  float2 p[PL_PTS_PER_LANE];
#pragma unroll
  for (int j = 0; j < PL_PTS_PER_LANE; ++j)
    p[j] = pts_lds[lane + j * 32];

  const int bd = s >> 3;   // b*D + d
  const int c  = s & 7;
  // out[bd][k][c*T + t], k-stride = C*T
  float* out0 = out + (size_t)bd * (2 * PL_C * PL_T) + c * PL_T;

  // ---- Wave w owns t = w, w+8, ... : all data already in registers -----
  for (int t = wave; t < PL_T; t += 8) {
    const float tt = (float)t * (1.0f / (float)(PL_T - 1));
    float m1 = 0.0f, m2 = 0.0f;   // triangle values are >= 0
#pragma unroll
    for (int j = 0; j < PL_PTS_PER_LANE; ++j) {
      float v  = fminf(tt - p[j].x, p[j].y - tt);
      v        = fmaxf(v, 0.0f);
      float hi = fmaxf(m1, v);
      float lo = fminf(m1, v);
      m1 = hi;
      m2 = fmaxf(m2, lo);
    }
    // wave32 top-2 merge: 5 shuffle-xor levels
#pragma unroll
    for (int off = 16; off > 0; off >>= 1) {
      float o1 = __shfl_xor(m1, off, 32);
      float o2 = __shfl_xor(m2, off, 32);
      float hi = fmaxf(m1, o1);
      float lo = fminf(m1, o1);
      m1 = hi;
      m2 = fmaxf(fmaxf(m2, o2), lo);
    }
    if (lane == 0) {
      out0[t]               = m1;  // k = 0
      out0[PL_C * PL_T + t] = m2;  // k = 1
    }
  }
}

extern "C" void kernel_launch(void* const* d_in, const int* in_sizes, int n_in,
                              void* d_out, int out_size, void* d_ws, size_t ws_size,
                              hipStream_t stream) {
  const float* diag = (const float*)d_in[0];
  float* out = (float*)d_out;
  const int n_slices = in_sizes[0] / (PL_N * 2);   // 1024 for the reference shape
  pl_topk_kernel<<<n_slices, 256, 0, stream>>>(diag, out);
}